// GlobalPool5_16784732193370
// MI455X (gfx1250) — compile-verified
//
#include <hip/hip_runtime.h>
#include <stdint.h>

#ifndef __has_builtin
#define __has_builtin(x) 0
#endif

#define DCH 128                 // channels (hardcoded in reference)
#define VEC (DCH / 4)           // 32 float4 per row; one per lane (wave32)
#define KTOP 3
#define TILE_NODES 8            // 8 rows * 512 B = 4 KB per tile
#define NBUF 2                  // double buffer -> 8 KB LDS per workgroup

typedef int v4i __attribute__((ext_vector_type(4)));
typedef __attribute__((address_space(1))) v4i gv4i;   // global int4
typedef __attribute__((address_space(3))) v4i lv4i;   // LDS int4
typedef __attribute__((address_space(1))) char gchar;
typedef __attribute__((address_space(3))) char lchar;

// ---------------------------------------------------------------------------
// CDNA5 async global->LDS copy of one tile (TILE_NODES rows x 512 B).
// Each async b128 instruction copies 16 B per lane (512 B per wave = 1 row);
// eight instructions with byte offsets 0..3584 move the whole tile.
// Tracked with ASYNCcnt (in-order completion for loads).
// Builtin signature (from hipcc diagnostic): (int4 AS1* src, int4 AS3* dst,
// imm int offset, imm int cpol).
// ---------------------------------------------------------------------------
__device__ __forceinline__ void async_copy_tile(const void* gp, void* lp) {
#if __has_builtin(__builtin_amdgcn_global_load_async_to_lds_b128)
  gchar* g = (gchar*)gp;
  lchar* l = (lchar*)lp;
  __builtin_amdgcn_global_load_async_to_lds_b128((gv4i*)(g + 0),    (lv4i*)(l + 0),    0, 0);
  __builtin_amdgcn_global_load_async_to_lds_b128((gv4i*)(g + 512),  (lv4i*)(l + 512),  0, 0);
  __builtin_amdgcn_global_load_async_to_lds_b128((gv4i*)(g + 1024), (lv4i*)(l + 1024), 0, 0);
  __builtin_amdgcn_global_load_async_to_lds_b128((gv4i*)(g + 1536), (lv4i*)(l + 1536), 0, 0);
  __builtin_amdgcn_global_load_async_to_lds_b128((gv4i*)(g + 2048), (lv4i*)(l + 2048), 0, 0);
  __builtin_amdgcn_global_load_async_to_lds_b128((gv4i*)(g + 2560), (lv4i*)(l + 2560), 0, 0);
  __builtin_amdgcn_global_load_async_to_lds_b128((gv4i*)(g + 3072), (lv4i*)(l + 3072), 0, 0);
  __builtin_amdgcn_global_load_async_to_lds_b128((gv4i*)(g + 3584), (lv4i*)(l + 3584), 0, 0);
#else
  uint32_t lds = (uint32_t)(uintptr_t)lp;   // addr[31:0] == wave-relative LDS byte address
  uint64_t ga  = (uint64_t)(uintptr_t)gp;
  asm volatile(
      "global_load_async_to_lds_b128 %0, %1, off\n\t"
      "global_load_async_to_lds_b128 %0, %1, off offset:512\n\t"
      "global_load_async_to_lds_b128 %0, %1, off offset:1024\n\t"
      "global_load_async_to_lds_b128 %0, %1, off offset:1536\n\t"
      "global_load_async_to_lds_b128 %0, %1, off offset:2048\n\t"
      "global_load_async_to_lds_b128 %0, %1, off offset:2560\n\t"
      "global_load_async_to_lds_b128 %0, %1, off offset:3072\n\t"
      "global_load_async_to_lds_b128 %0, %1, off offset:3584"
      :: "v"(lds), "v"(ga) : "memory");
#endif
}

// ASYNCcnt <= 8 (next tile may still be in flight) or == 0 (all done).
__device__ __forceinline__ void wait_async_le(bool keep8) {
#if __has_builtin(__builtin_amdgcn_s_wait_asynccnt)
  if (keep8) __builtin_amdgcn_s_wait_asynccnt(TILE_NODES);
  else       __builtin_amdgcn_s_wait_asynccnt(0);
#else
  if (keep8) asm volatile("s_wait_asynccnt 8" ::: "memory");
  else       asm volatile("s_wait_asynccnt 0" ::: "memory");
#endif
}

// DS and ASYNC queues are mutually unordered: drain our ds_loads of a buffer
// before the async engine is allowed to overwrite it.
__device__ __forceinline__ void wait_ds0() {
#if __has_builtin(__builtin_amdgcn_s_wait_dscnt)
  __builtin_amdgcn_s_wait_dscnt(0);
#else
  asm volatile("s_wait_dscnt 0" ::: "memory");
#endif
}

// ---------------------------------------------------------------------------
// One wave32 workgroup per graph. Lane t owns channels [4t, 4t+4) as float4.
// Lane 31's .w is channel 127 == the sort key; it tracks top-3 in registers.
// ---------------------------------------------------------------------------
__global__ __launch_bounds__(32) void pool5_kernel(const float* __restrict__ x,
                                                   const int* __restrict__ batch,
                                                   float* __restrict__ out,
                                                   int N) {
  __shared__ float4 tile[NBUF * TILE_NODES * VEC];  // 8 KB

  const int g = (int)blockIdx.x;
  const int lane = (int)threadIdx.x;

  // Segment bounds: batch is sorted ascending; binary search (uniform across
  // lanes -> broadcast loads, all hits in 192 MB L2 for the 8 MB batch array).
  int lo = 0, hi = N;
  while (lo < hi) { int m = (lo + hi) >> 1; if (batch[m] < g) lo = m + 1; else hi = m; }
  const int start = lo;
  hi = N;
  while (lo < hi) { int m = (lo + hi) >> 1; if (batch[m] <= g) lo = m + 1; else hi = m; }
  const int end = lo;
  const int count = end - start;

  float4 acc = make_float4(0.f, 0.f, 0.f, 0.f);
  const float NEG_INF = -__builtin_inff();
  float k0 = NEG_INF, k1 = NEG_INF, k2 = NEG_INF;   // meaningful on lane 31
  int   i0 = -1,      i1 = -1,      i2 = -1;

  const float4* __restrict__ xrow = reinterpret_cast<const float4*>(x);

  auto touch = [&](float4 v, int node) {
    acc.x += v.x; acc.y += v.y; acc.z += v.z; acc.w += v.w;
    const float key = v.w;  // channel 127 on lane 31; strict '>' keeps stability
    if (key > k0)      { k2 = k1; i2 = i1; k1 = k0; i1 = i0; k0 = key; i0 = node; }
    else if (key > k1) { k2 = k1; i2 = i1; k1 = key; i1 = node; }
    else if (key > k2) { k2 = key; i2 = node; }
  };

  const int ntiles = count / TILE_NODES;
  auto issue_tile = [&](int t, int buf) {
    const void* gp = (const void*)(x + ((size_t)(start + t * TILE_NODES) * DCH + lane * 4));
    void* lp = (void*)&tile[buf * TILE_NODES * VEC + lane];
    async_copy_tile(gp, lp);
  };

  // Prologue: fill the ring.
  int issued = 0;
  for (; issued < ntiles && issued < NBUF; ++issued) issue_tile(issued, issued & 1);

  // Steady state: wait for oldest tile, reduce it from LDS, refill its slot.
  for (int t = 0; t < ntiles; ++t) {
    wait_async_le(issued > t + 1);
    const float4* buf = &tile[(t & 1) * TILE_NODES * VEC];
#pragma unroll
    for (int j = 0; j < TILE_NODES; ++j)
      touch(buf[j * VEC + lane], start + t * TILE_NODES + j);
    if (issued < ntiles) { wait_ds0(); issue_tile(issued, issued & 1); ++issued; }
  }

  // Remainder (< TILE_NODES rows): direct global_load_b128 path.
  for (int node = start + ntiles * TILE_NODES; node < end; ++node)
    touch(xrow[(size_t)node * VEC + lane], node);

  // Broadcast lane 31's top-3 node indices to the wave (wave32 shuffle).
  i0 = __shfl(i0, 31);
  i1 = __shfl(i1, 31);
  i2 = __shfl(i2, 31);

  // Output row: [ mean(128) | sum(128) | top3 rows (3*128) ]
  float4* og = reinterpret_cast<float4*>(out + (size_t)g * (5 * DCH));
  const float inv = 1.0f / (float)(count > 0 ? count : 1);
  og[lane]       = make_float4(acc.x * inv, acc.y * inv, acc.z * inv, acc.w * inv);
  og[VEC + lane] = acc;

  const int idxs[KTOP] = { i0, i1, i2 };
#pragma unroll
  for (int j = 0; j < KTOP; ++j) {
    float4 v = make_float4(0.f, 0.f, 0.f, 0.f);
    if (idxs[j] >= 0) v = xrow[(size_t)idxs[j] * VEC + lane];  // L2-hot re-gather
    og[2 * VEC + j * VEC + lane] = v;
  }
}

extern "C" void kernel_launch(void* const* d_in, const int* in_sizes, int n_in,
                              void* d_out, int out_size, void* d_ws, size_t ws_size,
                              hipStream_t stream) {
  (void)n_in; (void)d_ws; (void)ws_size;
  const float* x   = (const float*)d_in[0];
  const int* batch = (const int*)d_in[1];
  float* out       = (float*)d_out;
  const int N = in_sizes[1];               // node count
  const int G = out_size / (5 * DCH);      // 8192 graphs
  pool5_kernel<<<dim3(G), dim3(32), 0, stream>>>(x, batch, out, N);
}